// AttentiveTransformer_38577396253271
// MI455X (gfx1250) — compile-verified
//
#include <hip/hip_runtime.h>
#include <hip/hip_bf16.h>

typedef float v2f __attribute__((ext_vector_type(2)));
typedef float v8f __attribute__((ext_vector_type(8)));

#define N_TOT  131072
#define NA     128
#define F_DIM  256
#define VBS    128
#define NCHUNK (N_TOT / VBS)   /* 1024 */
#define GAMMA  1.5f
#define EPS_BN 1e-5f

// ---------------------------------------------------------------------------
// Kernel 1: per ghost-batch chunk (128 rows):
//   x = a @ W.T + b   via v_wmma_f32_16x16x4_f32 (fp32-faithful WMMA path)
//   ghost-BN statistics -> folded (scale, shift) per (chunk, feature) in d_ws
//   raw x staged into d_out[0 .. N*F) (overwritten by m in kernel 2)
// Block: 256 threads = 8 wave32; wave w owns rows [16w, 16w+16) of the chunk.
// ---------------------------------------------------------------------------
__global__ __launch_bounds__(256) void k1_gemm_bnstats(
    const float* __restrict__ a,    // [N, NA]
    const float* __restrict__ W,    // [F, NA]
    const float* __restrict__ bvec, // [F]
    const float* __restrict__ bnw,  // [F]
    const float* __restrict__ bnb,  // [F]
    float* __restrict__ xout,       // [N, F]  (staging, = d_out first half)
    float* __restrict__ stats)      // [NCHUNK, 2, F]  scale | shift
{
    __shared__ float s_sum[F_DIM];
    __shared__ float s_sumsq[F_DIM];

    const int tid = threadIdx.x;
    if (tid < F_DIM) { s_sum[tid] = 0.0f; s_sumsq[tid] = 0.0f; }
    __syncthreads();

    const int chunk = blockIdx.x;
    const int wave  = tid >> 5;
    const int lane  = tid & 31;
    const int ml    = lane & 15;   // M (A) / N (B,C,D) position within tile
    const int h     = lane >> 4;   // half: selects K pair (A/B), M+8 (C/D)

    // A fragment source row for this lane (16x4 tile, M = ml, K = k0 + 2h + j)
    const int rowA = chunk * VBS + wave * 16 + ml;
    const float* __restrict__ arow = a + (size_t)rowA * NA;

    for (int n = 0; n < 16; ++n) {
        const int col = n * 16 + ml;                    // feature column (B: N = ml)
        const float* __restrict__ wrow = W + (size_t)col * NA;

        v8f acc = {0.f, 0.f, 0.f, 0.f, 0.f, 0.f, 0.f, 0.f};
        #pragma unroll 8
        for (int k0 = 0; k0 < NA; k0 += 4) {
            v2f af = *(const v2f*)(arow + k0 + 2 * h);  // A[M=ml][K=k0+2h+{0,1}]
            v2f bf = *(const v2f*)(wrow + k0 + 2 * h);  // B[K=k0+2h+{0,1}][N=ml]
            acc = __builtin_amdgcn_wmma_f32_16x16x4_f32(
                false, af, false, bf, (short)0, acc, false, false);
        }

        // add bias, accumulate BN partial sums, store x tile.
        const float bias = bvec[col];
        float psum = 0.0f, psq = 0.0f;
        #pragma unroll
        for (int v = 0; v < 8; ++v) {
            float x = acc[v] + bias;
            acc[v] = x;
            psum += x;
            psq  += x * x;
        }
        // D layout: VGPR v holds (M = v + 8h, N = ml)
        const int rbase = chunk * VBS + wave * 16 + 8 * h;
        #pragma unroll
        for (int v = 0; v < 8; ++v) {
            xout[(size_t)(rbase + v) * F_DIM + col] = acc[v];
        }
        // fold the two halves (16 rows of the strip) for the column sums
        psum += __shfl_xor(psum, 16, 32);
        psq  += __shfl_xor(psq, 16, 32);
        if (h == 0) {
            atomicAdd(&s_sum[col], psum);
            atomicAdd(&s_sumsq[col], psq);
        }
    }
    __syncthreads();

    if (tid < F_DIM) {
        const float inv_n = 1.0f / (float)VBS;
        float mean = s_sum[tid] * inv_n;
        float var  = s_sumsq[tid] * inv_n - mean * mean;   // biased var (torch BN)
        float inv  = rsqrtf(var + EPS_BN);
        float scale = bnw[tid] * inv;
        float shift = bnb[tid] - mean * scale;
        stats[(size_t)chunk * 2 * F_DIM + tid]         = scale;
        stats[(size_t)chunk * 2 * F_DIM + F_DIM + tid] = shift;
    }
}

// ---------------------------------------------------------------------------
// Kernel 2: one wave32 per row.
//   z = ((x * scale) + shift) * prior
//   repo-style sparsemax via rank counting (equivalent to ascending sort):
//     r_i  = #elements ordered before z_i, cs_i = sum of those + z_i
//     w_i  = (1 + r_i*z_i - cs_i) > 0 ; k_z = max(r_i * w_i)
//     tau  = (sum_{r<=k_z} z + 1) / k_z ; m = clip(z - tau, 0)
//   outputs: m (in place over x staging) and prior * (GAMMA - m).
// ---------------------------------------------------------------------------
__global__ __launch_bounds__(256) void k2_sparsemax(
    const float* __restrict__ xin,       // [N, F] staged x (= d_out first half)
    const float* __restrict__ prior,     // [N, F]
    const float* __restrict__ stats,     // [NCHUNK, 2, F]
    float* __restrict__ m_out,           // [N, F] (same buffer as xin)
    float* __restrict__ prior_out)       // [N, F]
{
    __shared__ float s_row[8][F_DIM];

    const int tid  = threadIdx.x;
    const int wave = tid >> 5;
    const int lane = tid & 31;
    const int row  = blockIdx.x * 8 + wave;
    const int chunk = row >> 7;                  // row / VBS
    const float* __restrict__ srow = stats + (size_t)chunk * 2 * F_DIM;

    float z[8], pr[8];
    #pragma unroll
    for (int e = 0; e < 8; ++e) {
        const int col = lane + 32 * e;
        float x = xin[(size_t)row * F_DIM + col];
        float p = prior[(size_t)row * F_DIM + col];
        float zz = (x * srow[col] + srow[F_DIM + col]) * p;
        z[e]  = zz;
        pr[e] = p;
        s_row[wave][col] = zz;
    }
    __syncthreads();

    // O(D^2) rank/prefix computation; ties broken by original index so the
    // result matches a stable ascending sort + cumsum.
    float rnk[8], slt[8];
    #pragma unroll
    for (int e = 0; e < 8; ++e) { rnk[e] = 0.0f; slt[e] = 0.0f; }

    for (int j = 0; j < F_DIM; ++j) {
        const float v = s_row[wave][j];          // LDS broadcast read
        #pragma unroll
        for (int e = 0; e < 8; ++e) {
            const int col = lane + 32 * e;
            bool lt = (v < z[e]) || ((v == z[e]) && (j < col));
            rnk[e] += lt ? 1.0f : 0.0f;
            slt[e] += lt ? v : 0.0f;
        }
    }

    // k_z = max(r * w); entries with w false contribute 0 (r=0 term is 0 anyway)
    float kz = 0.0f;
    #pragma unroll
    for (int e = 0; e < 8; ++e) {
        float cs = slt[e] + z[e];                // inclusive cumsum at rank r_i
        float cand = ((1.0f + rnk[e] * z[e] - cs) > 0.0f) ? rnk[e] : 0.0f;
        kz = fmaxf(kz, cand);
    }
    #pragma unroll
    for (int msk = 16; msk >= 1; msk >>= 1)
        kz = fmaxf(kz, __shfl_xor(kz, msk, 32));

    // m_z = sum of the first (k_z + 1) sorted entries
    float msum = 0.0f;
    #pragma unroll
    for (int e = 0; e < 8; ++e)
        msum += (rnk[e] <= kz) ? z[e] : 0.0f;
    #pragma unroll
    for (int msk = 16; msk >= 1; msk >>= 1)
        msum += __shfl_xor(msum, msk, 32);

    const float tau = (msum + 1.0f) / kz;        // kz==0 -> inf -> m == 0 (matches ref)

    #pragma unroll
    for (int e = 0; e < 8; ++e) {
        const int col = lane + 32 * e;
        float mm = fmaxf(z[e] - tau, 0.0f);
        m_out[(size_t)row * F_DIM + col]     = mm;
        prior_out[(size_t)row * F_DIM + col] = pr[e] * (GAMMA - mm);
    }
}

// ---------------------------------------------------------------------------
extern "C" void kernel_launch(void* const* d_in, const int* in_sizes, int n_in,
                              void* d_out, int out_size, void* d_ws, size_t ws_size,
                              hipStream_t stream) {
    const float* a     = (const float*)d_in[0];   // [N, NA]
    const float* prior = (const float*)d_in[1];   // [N, F]
    const float* W     = (const float*)d_in[2];   // [F, NA]
    const float* bvec  = (const float*)d_in[3];   // [F]
    const float* bnw   = (const float*)d_in[4];   // [F]
    const float* bnb   = (const float*)d_in[5];   // [F]

    float* out       = (float*)d_out;
    float* m_out     = out;                               // [N, F] (also x staging)
    float* prior_out = out + (size_t)N_TOT * F_DIM;       // [N, F]
    float* stats     = (float*)d_ws;                      // NCHUNK * 2 * F floats = 2 MB

    k1_gemm_bnstats<<<NCHUNK, 256, 0, stream>>>(a, W, bvec, bnw, bnb, m_out, stats);
    k2_sparsemax<<<N_TOT / 8, 256, 0, stream>>>(m_out, prior, stats, m_out, prior_out);
}